// RNNTJointNet_12635793785610
// MI455X (gfx1250) — compile-verified
//
#include <hip/hip_runtime.h>

typedef __attribute__((ext_vector_type(2))) float v2f;
typedef __attribute__((ext_vector_type(4))) float v4f;
typedef __attribute__((ext_vector_type(8))) float v8f;

constexpr int T = 256, B = 8, H1 = 512, U = 64, H2 = 512, V = 128;

// P[r][n] = sum_k A[r][k] * W[(h_off+k)*V + n]  (+ bias[n] if bias != null)
// mode 0 (pf): A[r][k] = f[(r%T)*B*H1 + (r/T)*H1 + k]   (M = B*T, row-reordered t,b -> b,t)
// mode 1 (pg): A[r][k] = g[r*H2 + k]                     (M = B*U, contiguous)
// One wave computes one 16x16 output tile via V_WMMA_F32_16X16X4_F32, K stepped by 4.
__global__ void __launch_bounds__(256)
joint_gemm_wmma(const float* __restrict__ A, const float* __restrict__ W,
                const float* __restrict__ bias, float* __restrict__ P,
                int K, int h_off, int mode)
{
    const int wave  = threadIdx.x >> 5;     // 0..7 -> N tile (V/16 = 8 tiles)
    const int lane  = threadIdx.x & 31;
    const int mt    = blockIdx.x;           // M tile
    const int l16   = lane & 15;
    const int hi    = lane >> 4;            // 0: K{0,1}, 1: K{2,3}
    const int khalf = hi << 1;
    const int n     = wave * 16 + l16;      // output column
    const int mrow  = mt * 16 + l16;        // A row for this lane

    size_t arow;
    if (mode == 0) arow = (size_t)(mrow % T) * (B * H1) + (size_t)(mrow / T) * H1;
    else           arow = (size_t)mrow * H2;

    const float* __restrict__ ap = A + arow + khalf;                 // 8B aligned
    const float* __restrict__ wp = W + (size_t)(h_off + khalf) * V + n;

    v8f c = {};
    #pragma unroll 4
    for (int k0 = 0; k0 < K; k0 += 4) {
        v2f a = *(const v2f*)(ap + k0);          // A[m][k0+khalf .. +1]
        v2f bv;
        bv.x = wp[(size_t)k0 * V];               // B[k0+khalf][n]
        bv.y = wp[(size_t)k0 * V + V];           // B[k0+khalf+1][n]
        c = __builtin_amdgcn_wmma_f32_16x16x4_f32(false, a, false, bv,
                                                  (short)0, c, false, false);
    }

    const float bb = bias ? bias[n] : 0.0f;
    const int mbase = mt * 16 + (hi << 3);       // VGPR j -> M = mbase + j
    #pragma unroll
    for (int j = 0; j < 8; ++j)
        P[(size_t)(mbase + j) * V + n] = c[j] + bb;
}

// h[b,t,u,:] = pf[b,t,:] + pg[b,u,:]   (bias already folded into pg)
// One wave per (b,t,u) row: 32 lanes x float4 = full V=128 row, 512B contiguous.
// Output is write-once -> non-temporal stores; pf/pg rows hit L0/L2.
__global__ void __launch_bounds__(256)
joint_add_bcast(const float* __restrict__ pf, const float* __restrict__ pg,
                float* __restrict__ out)
{
    const int tid  = blockIdx.x * 256 + threadIdx.x;
    const int lane = tid & 31;          // float4 index within the V row
    const int w    = tid >> 5;          // flat (b,t,u), u fastest
    const int u    = w & (U - 1);
    const int t    = (w >> 6) & (T - 1);
    const int b    = w >> 14;

    const v4f pfq = *(const v4f*)(pf + (size_t)(b * T + t) * V + lane * 4);
    const v4f pgq = *(const v4f*)(pg + (size_t)(b * U + u) * V + lane * 4);
    const v4f r = pfq + pgq;
    v4f* dst = (v4f*)(out + (size_t)w * V) + lane;
    __builtin_nontemporal_store(r, dst);
}

// Bit-copy f_lens (int32) into the tail of d_out.
__global__ void copy_lens(const int* __restrict__ lens, int* __restrict__ dst)
{
    if (threadIdx.x < B) dst[threadIdx.x] = lens[threadIdx.x];
}

extern "C" void kernel_launch(void* const* d_in, const int* in_sizes, int n_in,
                              void* d_out, int out_size, void* d_ws, size_t ws_size,
                              hipStream_t stream)
{
    const float* f      = (const float*)d_in[0];
    const int*   f_lens = (const int*)d_in[1];
    const float* g      = (const float*)d_in[2];
    // d_in[3] = g_lens (unused by reference output)
    const float* W      = (const float*)d_in[4];
    const float* bias   = (const float*)d_in[5];
    float* out = (float*)d_out;

    float* pf = (float*)d_ws;                  // B*T*V f32 = 1 MiB
    float* pg = pf + (size_t)B * T * V;        // B*U*V f32 = 256 KiB

    // pf = f @ Wf : M = B*T = 2048 -> 128 M-tiles, 8 waves/block cover N
    joint_gemm_wmma<<<(B * T) / 16, 256, 0, stream>>>(f, W, nullptr, pf, H1, 0, 0);
    // pg = g @ Wg + bias : M = B*U = 512 -> 32 M-tiles
    joint_gemm_wmma<<<(B * U) / 16, 256, 0, stream>>>(g, W, bias, pg, H2, H1, 1);

    // streaming broadcast add: B*T*U waves, one V-row each
    const int totalThreads = B * T * U * (V / 4);   // 4,194,304
    joint_add_bcast<<<totalThreads / 256, 256, 0, stream>>>(pf, pg, out);

    // tuple tail: f_lens
    copy_lens<<<1, 32, 0, stream>>>(f_lens, (int*)d_out + (size_t)B * T * U * V);
}